// Model_Multi_22419729285703
// MI455X (gfx1250) — compile-verified
//
#include <hip/hip_runtime.h>
#include <hip/hip_bf16.h>
#include <math.h>

typedef __attribute__((ext_vector_type(2))) float v2f;
typedef __attribute__((ext_vector_type(8))) float v8f;

// K-pair interleaved LDS layout: element (k, col) lives at
//   [(k>>1)*KPSTRIDE + col*2 + (k&1)]
// so the B fragment {(k,col),(k+1,col)} is one aligned ds_load_b64.
// KPSTRIDE = 288 dwords: kpair+1 is offset 288 = 32 (mod 64) banks, so
// lanes 0-15 (kpair p) use banks 0-31 and lanes 16-31 (kpair p+1) use
// banks 32-63 -> conflict-free b64 reads across the wave.
#define KPSTRIDE 288

// ---------------------------------------------------------------- utilities

__global__ __launch_bounds__(256) void zero_f32(float* __restrict__ p, long n) {
  long i = (long)blockIdx.x * 256 + threadIdx.x;
  long stride = (long)gridDim.x * 256;
  for (; i < n; i += stride) p[i] = 0.0f;
}

__global__ __launch_bounds__(256) void deg_init(float* __restrict__ deg, int n) {
  int i = blockIdx.x * 256 + threadIdx.x;
  if (i < n) deg[i] = 1.0f;  // self loop
}

__global__ __launch_bounds__(256) void deg_accum(const int* __restrict__ dst,
                                                 float* __restrict__ deg, int e) {
  int i = blockIdx.x * 256 + threadIdx.x;
  if (i < e) atomicAdd(&deg[dst[i]], 1.0f);
}

__global__ __launch_bounds__(256) void deg_to_dis(float* __restrict__ deg, int n) {
  int i = blockIdx.x * 256 + threadIdx.x;
  if (i < n) {
    float d = deg[i];
    deg[i] = rsqrtf(d < 1.0f ? 1.0f : d);
  }
}

// ------------------------------------------------- fp32 WMMA GEMM: X[N,128] @ W[128,128]
// 256 threads = 8 waves; block covers 128 rows; each wave: 16 rows x 128 cols
// = 8 tiles of 16x16 f32 accumulators; K-loop of 32 x V_WMMA_F32_16X16X4_F32.

__global__ __launch_bounds__(256) void gemm128_wmma(
    const float* __restrict__ X, const float* __restrict__ W,
    float* __restrict__ Out, int nrows, int accumulate)
{
  __shared__ float Wl[64 * KPSTRIDE];  // 72KB of the 320KB WGP LDS
  for (int idx = threadIdx.x; idx < 128 * 128; idx += 256) {
    int k = idx >> 7, col = idx & 127;
    Wl[(k >> 1) * KPSTRIDE + col * 2 + (k & 1)] = W[idx];
  }
  __syncthreads();

  const int wave = threadIdx.x >> 5;
  const int lane = threadIdx.x & 31;
  const int mrow = lane & 15;           // A row / B col within tile
  const int kho  = lane >> 4;           // K half: kpair offset 0 or 1
  const long row0 = (long)blockIdx.x * 128 + wave * 16;
  long rsrc = row0 + mrow;
  if (rsrc >= nrows) rsrc = nrows - 1;  // clamp (stores are guarded)
  const float* __restrict__ xr = X + rsrc * 128;

  v8f acc[8] = {};
#pragma unroll 4
  for (int k0 = 0; k0 < 128; k0 += 4) {
    const v2f a = *(const v2f*)(xr + k0 + kho * 2);
    const float* __restrict__ wrow = &Wl[((k0 >> 1) + kho) * KPSTRIDE];
#pragma unroll
    for (int t = 0; t < 8; ++t) {
      const v2f b = *(const v2f*)(wrow + (t * 16 + mrow) * 2);
      acc[t] = __builtin_amdgcn_wmma_f32_16x16x4_f32(
          false, a, false, b, (short)0, acc[t], false, false);
    }
  }

  const int mo = (lane >> 4) * 8;  // C layout: vgpr j -> M=j (lanes 0-15) / M=j+8
  const int nc = lane & 15;
#pragma unroll
  for (int t = 0; t < 8; ++t) {
#pragma unroll
    for (int j = 0; j < 8; ++j) {
      long r = row0 + mo + j;
      if (r < nrows) {
        float* p = Out + r * 128 + (long)(t * 16 + nc);
        float v = acc[t][j];
        *p = accumulate ? (*p + v) : v;
      }
    }
  }
}

// ------------------------------------------------- edge scatter: Agg[dst] += H[src]*coef

__global__ __launch_bounds__(256) void edge_aggregate(
    const float* __restrict__ H, const int* __restrict__ src,
    const int* __restrict__ dst, const float* __restrict__ dis,
    float* __restrict__ Agg, int e)
{
  long tid = (long)blockIdx.x * 256 + threadIdx.x;
  long ed = tid >> 5;                 // 32 lanes per edge, float4 per lane
  if (ed >= e) return;
  int s = src[ed], d = dst[ed];
  float c = dis[s] * dis[d];
  int j = (int)(tid & 31) * 4;
  const float4 h4 = *(const float4*)(H + (size_t)s * 128 + j);
  float* ap = Agg + (size_t)d * 128 + j;
  atomicAdd(ap + 0, h4.x * c);
  atomicAdd(ap + 1, h4.y * c);
  atomicAdd(ap + 2, h4.z * c);
  atomicAdd(ap + 3, h4.w * c);
}

// -------------------------------------- finalize: out = act(Agg + H*dis^2 + b)

__global__ __launch_bounds__(256) void conv_finalize(
    const float* __restrict__ Agg, const float* __restrict__ H,
    const float* __restrict__ dis, const float* __restrict__ bias,
    float* __restrict__ Out, int n, int act)
{
  long tid = (long)blockIdx.x * 256 + threadIdx.x;
  long total = (long)n * 128;
  if (tid >= total) return;
  long i = tid >> 7;
  int j = (int)(tid & 127);
  float di = dis[i];
  float v = Agg[tid] + H[tid] * di * di + bias[j];
  if (act == 1) v = v > 0.0f ? v : 0.0f;
  else if (act == 2) v = tanhf(v);
  Out[tid] = v;
}

// ------------------- finalize + LayerNorm fused (last conv), out -> d_out+1

__global__ __launch_bounds__(128) void conv_finalize_ln(
    const float* __restrict__ Agg, const float* __restrict__ H,
    const float* __restrict__ dis, const float* __restrict__ bias,
    const float* __restrict__ ln_g, const float* __restrict__ ln_b,
    float* __restrict__ Out, int n)
{
  int i = blockIdx.x;
  int j = threadIdx.x;
  if (i >= n) return;
  float di = dis[i];
  long idx = (long)i * 128 + j;
  float v = Agg[idx] + H[idx] * di * di + bias[j];
  __shared__ float s[128];
  s[j] = v; __syncthreads();
  for (int o = 64; o > 0; o >>= 1) { if (j < o) s[j] += s[j + o]; __syncthreads(); }
  float mu = s[0] * (1.0f / 128.0f);
  __syncthreads();
  float dv = v - mu;
  s[j] = dv * dv; __syncthreads();
  for (int o = 64; o > 0; o >>= 1) { if (j < o) s[j] += s[j + o]; __syncthreads(); }
  float var = s[0] * (1.0f / 128.0f);
  Out[idx] = dv * rsqrtf(var + 1e-5f) * ln_g[j] + ln_b[j];
}

// ------------------------------------------- CKA part 1: column sums of env/inv

__global__ __launch_bounds__(256) void colsum2(
    const float* __restrict__ X, const float* __restrict__ Y,
    float* __restrict__ Sx, float* __restrict__ Sy, int n, int rows_per_block)
{
  int j = threadIdx.x & 127;
  int which = threadIdx.x >> 7;
  const float* P = which ? Y : X;
  float* S = which ? Sy : Sx;
  long i0 = (long)blockIdx.x * rows_per_block;
  long i1 = i0 + rows_per_block; if (i1 > n) i1 = n;
  float acc = 0.0f;
  for (long i = i0; i < i1; ++i) acc += P[i * 128 + j];
  atomicAdd(&S[j], acc);
}

// -------- CKA part 2: split-K WMMA Gram. blockIdx.y: 0->X^T Y, 1->X^T X, 2->Y^T Y
// Stages 64-row sub-chunks of A and B in LDS (paired layout) so both fragments
// are single ds_load_b64, then runs the same DS->WMMA pipeline as the GEMM.

#define GRAM_CHUNK 2048
#define SUB 64

__global__ __launch_bounds__(256) void gram_wmma(
    const float* __restrict__ Xm, const float* __restrict__ Ym,
    float* __restrict__ G, int n)
{
  const float* A = (blockIdx.y == 2) ? Ym : Xm;
  const float* B = (blockIdx.y == 1) ? Xm : Ym;
  float* Gout = G + (size_t)blockIdx.y * 16384;

  __shared__ float Al[(SUB / 2) * KPSTRIDE];  // 36KB
  __shared__ float Bl[(SUB / 2) * KPSTRIDE];  // 36KB

  const int wave = threadIdx.x >> 5;
  const int lane = threadIdx.x & 31;
  const int mrow = lane & 15;
  const int kho  = lane >> 4;
  const int acol = wave * 16 + mrow;   // A^T row = column of env/inv

  long i0 = (long)blockIdx.x * GRAM_CHUNK;
  long i1 = i0 + GRAM_CHUNK; if (i1 > n) i1 = n;

  v8f acc[8] = {};
  for (long c0 = i0; c0 < i1; c0 += SUB) {
    __syncthreads();
    // cooperative stage: 64 rows x 128 cols of A and B, zero-filled tail
    for (int idx = threadIdx.x; idx < SUB * 128; idx += 256) {
      int k = idx >> 7, col = idx & 127;
      long row = c0 + k;
      int dst = (k >> 1) * KPSTRIDE + col * 2 + (k & 1);
      Al[dst] = (row < n) ? A[row * 128 + col] : 0.0f;
      Bl[dst] = (row < n) ? B[row * 128 + col] : 0.0f;
    }
    __syncthreads();
#pragma unroll 4
    for (int k0 = 0; k0 < SUB; k0 += 4) {
      const int kp = (k0 >> 1) + kho;
      const v2f a = *(const v2f*)(&Al[kp * KPSTRIDE + acol * 2]);
      const float* __restrict__ brow = &Bl[kp * KPSTRIDE];
#pragma unroll
      for (int t = 0; t < 8; ++t) {
        const v2f b = *(const v2f*)(brow + (t * 16 + mrow) * 2);
        acc[t] = __builtin_amdgcn_wmma_f32_16x16x4_f32(
            false, a, false, b, (short)0, acc[t], false, false);
      }
    }
  }

  const int mo = (lane >> 4) * 8;
  const int nc = lane & 15;
#pragma unroll
  for (int t = 0; t < 8; ++t) {
#pragma unroll
    for (int j = 0; j < 8; ++j) {
      atomicAdd(&Gout[(size_t)(wave * 16 + mo + j) * 128 + t * 16 + nc], acc[t][j]);
    }
  }
}

// -------- CKA part 3: hsic / (nx*ny) from raw moments; centered = S2 - s s^T / N

__global__ __launch_bounds__(256) void cka_final(
    const float* __restrict__ G, const float* __restrict__ Sx,
    const float* __restrict__ Sy, float* __restrict__ out, int n)
{
  float invN = 1.0f / (float)n;
  float h = 0.0f, xx = 0.0f, yy = 0.0f;
  for (int idx = threadIdx.x; idx < 16384; idx += 256) {
    int a = idx >> 7, b = idx & 127;
    float cxy = G[idx]         - Sx[a] * Sy[b] * invN;
    float cxx = G[16384 + idx] - Sx[a] * Sx[b] * invN;
    float cyy = G[32768 + idx] - Sy[a] * Sy[b] * invN;
    h  += cxy * cxy;
    xx += cxx * cxx;
    yy += cyy * cyy;
  }
  __shared__ float sh[256], sxx[256], syy[256];
  sh[threadIdx.x] = h; sxx[threadIdx.x] = xx; syy[threadIdx.x] = yy;
  __syncthreads();
  for (int o = 128; o > 0; o >>= 1) {
    if (threadIdx.x < o) {
      sh[threadIdx.x]  += sh[threadIdx.x + o];
      sxx[threadIdx.x] += sxx[threadIdx.x + o];
      syy[threadIdx.x] += syy[threadIdx.x + o];
    }
    __syncthreads();
  }
  if (threadIdx.x == 0) out[0] = sh[0] / (sqrtf(sxx[0]) * sqrtf(syy[0]));
}

// ---------------------------------------------------------------- launcher

extern "C" void kernel_launch(void* const* d_in, const int* in_sizes, int n_in,
                              void* d_out, int out_size, void* d_ws, size_t ws_size,
                              hipStream_t stream)
{
  (void)n_in; (void)out_size; (void)ws_size;

  const float* x     = (const float*)d_in[0];
  const int*   eidx  = (const int*)d_in[1];
  const float* W_ir1 = (const float*)d_in[2];  const float* b_ir1 = (const float*)d_in[3];
  const float* W_ir2 = (const float*)d_in[4];  const float* b_ir2 = (const float*)d_in[5];
  const float* W_g1  = (const float*)d_in[6];  const float* b_g1  = (const float*)d_in[7];
  const float* W_g2  = (const float*)d_in[8];  const float* b_g2  = (const float*)d_in[9];
  const float* W_d1  = (const float*)d_in[10]; const float* b_d1  = (const float*)d_in[11];
  const float* W_d2  = (const float*)d_in[12]; const float* b_d2  = (const float*)d_in[13];
  const float* ln_g  = (const float*)d_in[14]; const float* ln_b  = (const float*)d_in[15];

  const int n = in_sizes[0] / 128;
  const int e = in_sizes[1] / 2;
  const int* srcp = eidx;
  const int* dstp = eidx + e;

  float* out = (float*)d_out;            // out[0]=ind_loss, out+1 = rebuilt[N,128]
  float* ws  = (float*)d_ws;
  const size_t NF = (size_t)n * 128;
  float* dis  = ws;                      // N
  float* bufA = ws + (((size_t)n + 3) & ~(size_t)3);
  float* bufB = bufA + NF;
  float* Hb   = bufB + NF;
  float* env  = Hb + NF;
  float* invb = env + NF;
  float* Sx   = invb + NF;               // 128
  float* Sy   = Sx + 128;                // 128
  float* G    = Sy + 128;                // 3 * 128*128

  const int nodeBlk  = (n + 255) / 256;
  const int edgeBlk  = (e + 255) / 256;
  const int gemmBlk  = (n + 127) / 128;
  const int elemBlk  = (int)((NF + 255) / 256);
  const int aggBlk   = (int)(((long)e * 32 + 255) / 256);

  // degree / normalization coefficients: dis = rsqrt(1 + indeg)
  deg_init<<<nodeBlk, 256, 0, stream>>>(dis, n);
  deg_accum<<<edgeBlk, 256, 0, stream>>>(dstp, dis, e);
  deg_to_dis<<<nodeBlk, 256, 0, stream>>>(dis, n);

  auto conv = [&](const float* in, const float* W, const float* bias,
                  float* agg, float* outp, int act) {
    gemm128_wmma<<<gemmBlk, 256, 0, stream>>>(in, W, Hb, n, 0);
    zero_f32<<<elemBlk, 256, 0, stream>>>(agg, (long)NF);
    edge_aggregate<<<aggBlk, 256, 0, stream>>>(Hb, srcp, dstp, dis, agg, e);
    conv_finalize<<<elemBlk, 256, 0, stream>>>(agg, Hb, dis, bias, outp, n, act);
  };

  // ir_Learner: env = tanh(conv(relu(conv(x))))
  conv(x,    W_ir1, b_ir1, bufA, bufA, /*relu*/1);
  conv(bufA, W_ir2, b_ir2, bufB, env,  /*tanh*/2);
  // main gnn: inv = conv(relu(conv(x)))
  conv(x,    W_g1,  b_g1,  bufA, bufA, 1);
  conv(bufA, W_g2,  b_g2,  bufB, invb, 0);

  // decoder conv1 on concat([env, inv]) == env@Wd1_top + inv@Wd1_bot
  gemm128_wmma<<<gemmBlk, 256, 0, stream>>>(env,  W_d1,          Hb, n, 0);
  gemm128_wmma<<<gemmBlk, 256, 0, stream>>>(invb, W_d1 + 16384,  Hb, n, 1);
  zero_f32<<<elemBlk, 256, 0, stream>>>(bufA, (long)NF);
  edge_aggregate<<<aggBlk, 256, 0, stream>>>(Hb, srcp, dstp, dis, bufA, e);
  conv_finalize<<<elemBlk, 256, 0, stream>>>(bufA, Hb, dis, b_d1, bufA, n, 1);

  // decoder conv2 + fused LayerNorm -> rebuilt
  gemm128_wmma<<<gemmBlk, 256, 0, stream>>>(bufA, W_d2, Hb, n, 0);
  zero_f32<<<elemBlk, 256, 0, stream>>>(bufB, (long)NF);
  edge_aggregate<<<aggBlk, 256, 0, stream>>>(Hb, srcp, dstp, dis, bufB, e);
  conv_finalize_ln<<<n, 128, 0, stream>>>(bufB, Hb, dis, b_d2, ln_g, ln_b, out + 1, n);

  // linear CKA(env, inv)
  zero_f32<<<194, 256, 0, stream>>>(Sx, 256 + 3 * 16384);
  colsum2<<<(n + 511) / 512, 256, 0, stream>>>(env, invb, Sx, Sy, n, 512);
  dim3 ggrid((n + GRAM_CHUNK - 1) / GRAM_CHUNK, 3);
  gram_wmma<<<ggrid, 256, 0, stream>>>(env, invb, G, n);
  cka_final<<<1, 256, 0, stream>>>(G, Sx, Sy, out, n);
}